// AttentionLayer_63256278336133
// MI455X (gfx1250) — compile-verified
//
#include <hip/hip_runtime.h>
#include <stdint.h>

#define WINDOW    5
#define TW        11          // 2*WINDOW+1
#define MAXCS     166
#define EMB       768
#define SEQ       1126
#define NBATCH    16
#define CHUNK     16          // s-positions per workgroup
#define NBUF      3           // triple-buffered TDM pipeline
#define ROWSTRIDE (2 * EMB)   // data row stride in elements (LM dimension = 2)

typedef unsigned int u32x4 __attribute__((ext_vector_type(4)));
typedef int          i32x8 __attribute__((ext_vector_type(8)));
typedef int          i32x4 __attribute__((ext_vector_type(4)));

// Issue one TDM gather: 11 rows (clipped window indices) x 768 floats from the
// snippet's LM_IDX=0 plane into LDS at lds_off. Descriptor per CDNA5 ISA §8.
__device__ __forceinline__ void tdm_gather(unsigned lds_off,
                                           unsigned long long gaddr,
                                           int inp) {
  int id[TW];
#pragma unroll
  for (int k = 0; k < TW; ++k) {
    int v = inp + (k - WINDOW);
    v = v < 0 ? 0 : v;
    v = v > (MAXCS - 1) ? (MAXCS - 1) : v;
    id[k] = v;  // non-decreasing after clip -> OOB-safe gather ordering
  }

  u32x4 g0;
  // count=1 | gather_index_size=0 (16-bit) | gather_mode=1
  g0[0] = 0x80000001u;
  g0[1] = lds_off;                                   // lds_addr [63:32]
  g0[2] = (unsigned)(gaddr & 0xFFFFFFFFull);         // global_addr [95:64]
  g0[3] = (unsigned)(gaddr >> 32) | 0x80000000u;     // global_addr hi | type=2<<30

  i32x8 g1;
  g1[0] = 0x00020000;                               // data_size=2 (4 bytes)
  g1[1] = (int)((ROWSTRIDE & 0xFFFF) << 16);        // tensor_dim0 lo16 (=1536)
  g1[2] = (int)((ROWSTRIDE >> 16) | (MAXCS << 16)); // dim0 hi | tensor_dim1 lo (=166)
  g1[3] = (int)(EMB << 16);                         // dim1 hi | tile_dim0 (=768)
  g1[4] = TW;                                       // tile_dim1 = #valid indices
  g1[5] = ROWSTRIDE;                                // tensor_dim0_stride lo32
  g1[6] = 0;
  g1[7] = 0;

  i32x4 g2, g3;                                     // 16-bit row indices
  g2[0] = id[0] | (id[1] << 16);
  g2[1] = id[2] | (id[3] << 16);
  g2[2] = id[4] | (id[5] << 16);
  g2[3] = id[6] | (id[7] << 16);
  g3[0] = id[8] | (id[9] << 16);
  g3[1] = id[10];
  g3[2] = 0;
  g3[3] = 0;

#if __clang_major__ >= 23
  i32x8 gz = {0, 0, 0, 0, 0, 0, 0, 0};
  __builtin_amdgcn_tensor_load_to_lds(g0, g1, g2, g3, gz, 0);
#else
  __builtin_amdgcn_tensor_load_to_lds(g0, g1, g2, g3, 0);
#endif
}

__global__ __launch_bounds__(256) void window_gather_dot_kernel(
    const int* __restrict__ inputs,            // (B, SEQ)
    const int* __restrict__ code_snippet_id,   // (B, 1)
    const float* __restrict__ data,            // (100, 166, 2, 768)
    const float* __restrict__ w,               // (11, 768)
    float* __restrict__ out)                   // (B, SEQ, 768)
{
  __shared__ float sbuf[NBUF][TW * EMB];       // 3 x 33 KB gather buffers

  const int tid = (int)threadIdx.x;
  const int b   = (int)blockIdx.y;
  const int s0  = (int)blockIdx.x * CHUNK;
  int n = SEQ - s0;
  if (n > CHUNK) n = CHUNK;

  const int snip = code_snippet_id[b];
  const unsigned long long gbase =
      (unsigned long long)(uintptr_t)data +
      (unsigned long long)snip * (unsigned long long)(MAXCS * ROWSTRIDE) * 4ull;

  const unsigned lds_base   = (unsigned)(uintptr_t)&sbuf[0][0];
  const unsigned lds_stride = (unsigned)(TW * EMB * 4);

  // Scalar (per-wave-uniform) predicate so only wave 0 carries the TDM ops.
  const bool wave0 = (__builtin_amdgcn_readfirstlane(tid) < 32);

  // Kick off the first two gathers immediately; the w-register preamble below
  // overlaps with the DMA in flight.
  if (wave0) {
    tdm_gather(lds_base, gbase, inputs[b * SEQ + s0]);
    if (n > 1) tdm_gather(lds_base + lds_stride, gbase, inputs[b * SEQ + s0 + 1]);
  }

  // Hoist w into registers: each thread owns channels tid, tid+256, tid+512.
  float wr0[TW], wr1[TW], wr2[TW];
#pragma unroll
  for (int k = 0; k < TW; ++k) {
    wr0[k] = w[k * EMB + tid];
    wr1[k] = w[k * EMB + tid + 256];
    wr2[k] = w[k * EMB + tid + 512];
  }

  for (int i = 0; i < n; ++i) {
    const int ib = i % NBUF;
    if (wave0) {
      if (i + 2 < n) {
        // Issue two-ahead into the buffer freed at the end of iteration i-1,
        // then wait for the current buffer (in-order TENSORcnt: <=2 pending
        // means the DMA for buffer ib has retired).
        tdm_gather(lds_base + (unsigned)((i + 2) % NBUF) * lds_stride, gbase,
                   inputs[b * SEQ + s0 + i + 2]);
        __builtin_amdgcn_s_wait_tensorcnt(2);
      } else if (i + 1 < n) {
        __builtin_amdgcn_s_wait_tensorcnt(1);
      } else {
        __builtin_amdgcn_s_wait_tensorcnt(0);
      }
    }
    __syncthreads();  // current buffer DMA complete + visible to all waves

    const float* sb = &sbuf[ib][0];
    float a0 = 0.f, a1 = 0.f, a2 = 0.f;
#pragma unroll
    for (int k = 0; k < TW; ++k) {
      a0 = __builtin_fmaf(sb[k * EMB + tid],       wr0[k], a0);
      a1 = __builtin_fmaf(sb[k * EMB + tid + 256], wr1[k], a1);
      a2 = __builtin_fmaf(sb[k * EMB + tid + 512], wr2[k], a2);
    }
    const size_t ob = (size_t)(b * SEQ + s0 + i) * EMB;
    out[ob + tid]       = a0;
    out[ob + tid + 256] = a1;
    out[ob + tid + 512] = a2;

    __syncthreads();  // all reads of this buffer done before it is re-filled
  }
}

extern "C" void kernel_launch(void* const* d_in, const int* in_sizes, int n_in,
                              void* d_out, int out_size, void* d_ws,
                              size_t ws_size, hipStream_t stream) {
  (void)in_sizes; (void)n_in; (void)d_ws; (void)ws_size; (void)out_size;
  const int*   inputs = (const int*)d_in[0];
  const int*   snip   = (const int*)d_in[1];
  const float* data   = (const float*)d_in[2];
  const float* w      = (const float*)d_in[3];
  float*       out    = (float*)d_out;

  dim3 grid((SEQ + CHUNK - 1) / CHUNK, NBATCH);
  window_gather_dot_kernel<<<grid, 256, 0, stream>>>(inputs, snip, data, w, out);
}